// ClusterGNN_67723044323358
// MI455X (gfx1250) — compile-verified
//
#include <hip/hip_runtime.h>

typedef float v2f __attribute__((ext_vector_type(2)));
typedef float v8f __attribute__((ext_vector_type(8)));

// ---------------- degree / normalization ----------------
__global__ void k_init_deg(float* __restrict__ deg, int n) {
  int i = blockIdx.x * blockDim.x + threadIdx.x;
  if (i < n) deg[i] = 1.0f;  // self-loop contribution
}

__global__ void k_accum_deg(const long long* __restrict__ col, float* __restrict__ deg, int E) {
  int e = blockIdx.x * blockDim.x + threadIdx.x;
  if (e < E) atomicAdd(&deg[col[e]], 1.0f);
}

__global__ void k_deg_to_dinv(float* __restrict__ deg, int n) {
  int i = blockIdx.x * blockDim.x + threadIdx.x;
  if (i < n) {
    float d = deg[i];
    deg[i] = (d > 0.0f) ? rsqrtf(d) : 0.0f;
  }
}

// ---------------- fp32 WMMA GEMM: C[n x F] = A[n x K] * B[K x F] ----------------
// One wave per 16x16 C tile; block = F/16 waves sharing one 16xK A tile in LDS.
template <int K, int F>
__global__ void gemm_wmma_f32(const float* __restrict__ A, const float* __restrict__ B,
                              float* __restrict__ C, int nrows) {
  static_assert(K % 4 == 0 && F % 16 == 0, "shape");
  constexpr int LDK = K + 1;  // pad: stride K is a multiple of 64 banks
  __shared__ float lds_a[16 * LDK];

  const int m0   = blockIdx.x * 16;
  const int wave = threadIdx.x >> 5;
  const int lane = threadIdx.x & 31;
  const int half = lane >> 4;   // 0: lanes 0-15, 1: lanes 16-31
  const int lrow = lane & 15;

  // cooperative, coalesced load of the 16 x K A tile
  for (int i = threadIdx.x; i < 16 * K; i += blockDim.x) {
    int r = i / K, c = i % K;
    int rr = m0 + r; if (rr >= nrows) rr = nrows - 1;  // clamp (n is a multiple of 16 here)
    lds_a[r * LDK + c] = A[(long long)rr * K + c];
  }
  __syncthreads();

  const int n0 = wave * 16;
  v8f acc = {};
  for (int kb = 0; kb < K; kb += 4) {
    // A 16x4 frag: VGPR0 holds K=kb+2*half, VGPR1 holds K=kb+2*half+1, M = lane&15
    v2f a, b;
    a.x = lds_a[lrow * LDK + kb + 2 * half + 0];
    a.y = lds_a[lrow * LDK + kb + 2 * half + 1];
    // B 4x16 frag: VGPR v holds row K=kb+v (lanes 0-15) / K=kb+v+2 (lanes 16-31), N = lane&15
    b.x = B[(long long)(kb + 2 * half + 0) * F + n0 + lrow];
    b.y = B[(long long)(kb + 2 * half + 1) * F + n0 + lrow];
    acc = __builtin_amdgcn_wmma_f32_16x16x4_f32(false, a, false, b, (short)0, acc, false, false);
  }

  // C/D layout: VGPR j -> M = j + 8*half, N = lane&15
#pragma unroll
  for (int j = 0; j < 8; ++j) {
    int m = m0 + j + 8 * half;
    if (m < nrows) C[(long long)m * F + n0 + lrow] = acc[j];
  }
}

// ---------------- out[i][f] = b[f] + h[i][f] * dinv[i]^2 (self-loop + bias) ----------------
template <int F>
__global__ void k_self_bias(const float* __restrict__ h, const float* __restrict__ dinv,
                            const float* __restrict__ bias, float* __restrict__ out, int n) {
  constexpr int LOGF = (F == 64) ? 6 : 5;
  long long gid = (long long)blockIdx.x * blockDim.x + threadIdx.x;
  if (gid < (long long)n * F) {
    int i = (int)(gid >> LOGF);
    int f = (int)(gid & (F - 1));
    float di = dinv[i];
    out[gid] = bias[f] + h[gid] * di * di;
  }
}

// ---------------- scatter-add: out[c] += h[r] * dinv[r]*dinv[c] ----------------
template <int F>
__global__ void edge_aggregate(const long long* __restrict__ row, const long long* __restrict__ col,
                               const float* __restrict__ h, const float* __restrict__ dinv,
                               float* __restrict__ out, int E) {
  constexpr int LOGF = (F == 64) ? 6 : 5;
  long long gid = (long long)blockIdx.x * blockDim.x + threadIdx.x;
  if (gid >= (long long)E * F) return;
  long long e = gid >> LOGF;
  int f = (int)(gid & (F - 1));
  long long r = row[e], c = col[e];
  float w = dinv[r] * dinv[c];
  atomicAdd(&out[c * F + f], h[r * F + f] * w);
}

__global__ void k_relu(float* __restrict__ x, long long total) {
  long long gid = (long long)blockIdx.x * blockDim.x + threadIdx.x;
  if (gid < total) x[gid] = fmaxf(x[gid], 0.0f);
}

// ---------------- per-row log_softmax over 32 classes: one wave32 per row ----------------
__global__ void k_log_softmax32(float* __restrict__ out, int n) {
  int wid  = (int)(((long long)blockIdx.x * blockDim.x + threadIdx.x) >> 5);
  int lane = threadIdx.x & 31;
  if (wid >= n) return;
  long long idx = (long long)wid * 32 + lane;
  float v = out[idx];
  float m = v;
  for (int off = 16; off > 0; off >>= 1) m = fmaxf(m, __shfl_xor(m, off, 32));
  float ex = __expf(v - m);
  float s = ex;
  for (int off = 16; off > 0; off >>= 1) s += __shfl_xor(s, off, 32);
  out[idx] = v - m - __logf(s);
}

extern "C" void kernel_launch(void* const* d_in, const int* in_sizes, int n_in,
                              void* d_out, int out_size, void* d_ws, size_t ws_size,
                              hipStream_t stream) {
  const float*     x  = (const float*)d_in[0];
  const long long* ei = (const long long*)d_in[1];  // int64 edge_index [2, E]
  const float*     W1 = (const float*)d_in[2];
  const float*     b1 = (const float*)d_in[3];
  const float*     W2 = (const float*)d_in[4];
  const float*     b2 = (const float*)d_in[5];
  float* out = (float*)d_out;

  const int n = in_sizes[0] / 128;  // 100000 (multiple of 16)
  const int E = in_sizes[1] / 2;    // 1600000
  const long long* row = ei;
  const long long* col = ei + E;

  // workspace layout (floats): dinv[n] | h1[n*64] | g1[n*64] | h2[n*32]  (~64.4 MB)
  float* dinv = (float*)d_ws;
  float* h1   = dinv + n;
  float* g1   = h1 + (size_t)n * 64;
  float* h2   = g1 + (size_t)n * 64;

  const int T = 256;

  // --- degrees & symmetric normalization ---
  k_init_deg<<<(n + T - 1) / T, T, 0, stream>>>(dinv, n);
  k_accum_deg<<<(E + T - 1) / T, T, 0, stream>>>(col, dinv, E);
  k_deg_to_dinv<<<(n + T - 1) / T, T, 0, stream>>>(dinv, n);

  // --- layer 1: h1 = x @ W1 ; g1 = b1 + selfloop + scatter ; relu ---
  gemm_wmma_f32<128, 64><<<n / 16, 4 * 32, 0, stream>>>(x, W1, h1, n);
  {
    long long tot = (long long)n * 64;
    k_self_bias<64><<<(unsigned)((tot + T - 1) / T), T, 0, stream>>>(h1, dinv, b1, g1, n);
    long long etot = (long long)E * 64;
    edge_aggregate<64><<<(unsigned)((etot + T - 1) / T), T, 0, stream>>>(row, col, h1, dinv, g1, E);
    k_relu<<<(unsigned)((tot + T - 1) / T), T, 0, stream>>>(g1, tot);
  }

  // --- layer 2: h2 = g1 @ W2 ; out = b2 + selfloop + scatter ; log_softmax ---
  gemm_wmma_f32<64, 32><<<n / 16, 2 * 32, 0, stream>>>(g1, W2, h2, n);
  {
    long long tot = (long long)n * 32;
    k_self_bias<32><<<(unsigned)((tot + T - 1) / T), T, 0, stream>>>(h2, dinv, b2, out, n);
    long long etot = (long long)E * 32;
    edge_aggregate<32><<<(unsigned)((etot + T - 1) / T), T, 0, stream>>>(row, col, h2, dinv, out, E);
    k_log_softmax32<<<(unsigned)((tot + T - 1) / T), T, 0, stream>>>(out, n);
  }
}